// LabelsEncoder_80522046865453
// MI455X (gfx1250) — compile-verified
//
#include <hip/hip_runtime.h>

// Problem constants (match reference)
constexpr int VOCAB = 30522;
constexpr int D = 768;
constexpr int B = 4, Q = 16, W = 12, P = 64;
constexpr int THREADS = 192;            // 192 lanes * 4 floats = 768 = D (one row per block)

#if defined(__AMDGCN__) && \
    __has_builtin(__builtin_amdgcn_global_load_async_to_lds_b128) && \
    __has_builtin(__builtin_amdgcn_global_store_async_from_lds_b128)
#define USE_ASYNC_LDS 1
#else
#define USE_ASYNC_LDS 0
#endif

// Exact parameter type per hipcc diagnostic: int __attribute__((vector_size(16)))*
typedef int v4i __attribute__((vector_size(4 * sizeof(int))));
typedef __attribute__((address_space(1))) v4i gv4i;   // global
typedef __attribute__((address_space(3))) v4i lv4i;   // LDS

__device__ __forceinline__ void wait_asynccnt0() {
#if __has_builtin(__builtin_amdgcn_s_wait_asynccnt)
  __builtin_amdgcn_s_wait_asynccnt(0);
#else
  asm volatile("s_wait_asynccnt 0x0" ::: "memory");
#endif
}

__global__ __launch_bounds__(THREADS)
void labels_encoder_kernel(const int* __restrict__ queries,     // (B,Q,W)
                           const int* __restrict__ heads,       // (B,Q,1)
                           const int* __restrict__ labels,      // (B,P)
                           const float* __restrict__ proposals, // (B,P,4)
                           const float* __restrict__ emb,       // (VOCAB,D)
                           float* __restrict__ out) {           // (B,P,Q,D)
  const int r  = blockIdx.x;            // flat row over (b,p,q)
  const int b  = r / (P * Q);
  const int pq = r - b * (P * Q);
  const int p  = pq / Q;
  const int q  = pq - p * Q;
  const int tid = threadIdx.x;

  // ---- per-row scalar logic (block-uniform) ----
  const int* qrow = queries + (b * Q + q) * W;
  int qv[W];
  bool anyword = false;
#pragma unroll
  for (int w = 0; w < W; ++w) { qv[w] = qrow[w]; anyword |= (qv[w] != 0); }

  const int hq = heads[b * Q + q];

  // jnp.argmax(queries * is_head) — first occurrence of the maximum
  int pos   = 0;
  int bestv = (qv[0] == hq && anyword) ? qv[0] : 0;
#pragma unroll
  for (int w = 1; w < W; ++w) {
    const int m = (qv[w] == hq && anyword) ? qv[w] : 0;
    if (m > bestv) { bestv = m; pos = w; }
  }

  const int  qp         = qv[pos];
  const bool is_head_at = (qp == hq) && anyword;
  const int  token      = is_head_at ? labels[b * P + p] : qp;
  const bool is_word_at = (qp != 0);

  const float* pr = proposals + (b * P + p) * 4;
  const bool is_prop =
      (pr[0] != 0.f) || (pr[1] != 0.f) || (pr[2] != 0.f) || (pr[3] != 0.f);

  const bool keep = anyword && is_prop && is_word_at;

  float* orow = out + (size_t)r * D;

  if (!keep) {
    // Harness poisons d_out: must write zeros explicitly every call.
    *reinterpret_cast<float4*>(orow + tid * 4) = make_float4(0.f, 0.f, 0.f, 0.f);
    return;
  }

  const float* srow = emb + (size_t)token * D;
  // gfx1250 prefetch (lowers to global_prefetch_b8)
  __builtin_prefetch(srow + tid * 4, 0, 1);

#if USE_ASYNC_LDS
  // Pure DMA-style pipe: global -> LDS -> global, no VALU on the payload.
  // Each lane moves its own 16B chunk; waits are per-wave via ASYNCcnt.
  __shared__ float lds[D];
  gv4i* gsrc = (gv4i*)(srow + tid * 4);
  gv4i* gdst = (gv4i*)(orow + tid * 4);
  lv4i* lbuf = (lv4i*)(&lds[tid * 4]);

  __builtin_amdgcn_global_load_async_to_lds_b128(gsrc, lbuf, 0, 0);
  wait_asynccnt0();                      // LDS chunk written by this wave's load
  __builtin_amdgcn_global_store_async_from_lds_b128(gdst, lbuf, 0, 0);
  wait_asynccnt0();                      // drain before wave exit
#else
  // Fallback: wide B128 copy through VGPRs.
  const float4 v = *reinterpret_cast<const float4*>(srow + tid * 4);
  *reinterpret_cast<float4*>(orow + tid * 4) = v;
#endif
}

extern "C" void kernel_launch(void* const* d_in, const int* in_sizes, int n_in,
                              void* d_out, int out_size, void* d_ws, size_t ws_size,
                              hipStream_t stream) {
  const int*   queries   = (const int*)d_in[0];
  const int*   heads     = (const int*)d_in[1];
  const int*   labels    = (const int*)d_in[2];
  const float* proposals = (const float*)d_in[3];
  const float* emb       = (const float*)d_in[4];
  float*       out       = (float*)d_out;

  (void)in_sizes; (void)n_in; (void)out_size; (void)d_ws; (void)ws_size; (void)VOCAB;

  dim3 grid(B * P * Q);   // 4096 rows
  dim3 block(THREADS);    // 192 threads = 6 wave32s, one float4 per lane
  hipLaunchKernelGGL(labels_encoder_kernel, grid, block, 0, stream,
                     queries, heads, labels, proposals, emb, out);
}